// DynamicConv2d_5686536700035
// MI455X (gfx1250) — compile-verified
//
#include <hip/hip_runtime.h>
#include <math.h>

typedef float v2f __attribute__((ext_vector_type(2)));
typedef float v8f __attribute__((ext_vector_type(8)));

#define BN_EPS 1e-5f
// B=32, C=256, H=W=96, G=2, K=3, CR=64
// proj input matrix P: M_ROWS = 32*9 (pooled) + 32 (gap) = 320 rows, 256 cols

// ---------------------------------------------------------------------------
// Kernel 1: pooling. One block per (b,c) plane (8192 blocks).
// Produces P[m,c]: rows b*9+kk = pooled bins / 1024, rows 288+b = global mean.
// ---------------------------------------------------------------------------
__global__ __launch_bounds__(256) void pool_kernel(const float* __restrict__ x,
                                                   float* __restrict__ P) {
    __shared__ float bins[9 * 256];
    const int t = threadIdx.x;
    const int plane = blockIdx.x;                 // b*256 + c
    const float4* xp = (const float4*)(x + (size_t)plane * 9216);

#pragma unroll
    for (int i = 0; i < 9; ++i) bins[i * 256 + t] = 0.f;   // own column only

#pragma unroll
    for (int i = 0; i < 9; ++i) {
        const int g = t + i * 256;                // float4 index 0..2303, contiguous per iter
        const float4 v = xp[g];
        const int row = g / 24;                   // 24 float4 per 96-wide row
        const int cb  = (g % 24) >> 3;            // column block (32 floats = 8 float4)
        const int bin = (row >> 5) * 3 + cb;
        bins[bin * 256 + t] += v.x + v.y + v.z + v.w;
    }
    __syncthreads();
    for (int s = 128; s > 0; s >>= 1) {
        if (t < s) {
#pragma unroll
            for (int bn = 0; bn < 9; ++bn) bins[bn * 256 + t] += bins[bn * 256 + t + s];
        }
        __syncthreads();
    }
    const int b = plane >> 8, c = plane & 255;
    if (t < 9) P[(b * 9 + t) * 256 + c] = bins[t * 256] * (1.f / 1024.f);
    if (t == 0) {
        float s = 0.f;
#pragma unroll
        for (int bn = 0; bn < 9; ++bn) s += bins[bn * 256];
        P[(288 + b) * 256 + c] = s * (1.f / 9216.f);
    }
}

// ---------------------------------------------------------------------------
// Kernel 2: H = GELU(BN(P @ w1^T)).  M=320 (20 tiles), K=256, N=64 (4 tiles).
// One wave per 16x16 output tile, V_WMMA_F32_16X16X4_F32.
// ---------------------------------------------------------------------------
__global__ __launch_bounds__(128) void gemm1_kernel(const float* __restrict__ P,
                                                    const float* __restrict__ w1,
                                                    const float* __restrict__ gamma,
                                                    const float* __restrict__ beta,
                                                    const float* __restrict__ mean,
                                                    const float* __restrict__ var,
                                                    float* __restrict__ Hbuf) {
    const int lane  = threadIdx.x & 31;
    const int wv    = threadIdx.x >> 5;
    const int Mbase = blockIdx.x * 16;
    const int Nbase = wv * 16;
    const int rA    = Mbase + (lane & 15);
    const int nB    = Nbase + (lane & 15);
    const int koff  = (lane >> 4) << 1;           // 0 (lanes 0-15) or 2 (lanes 16-31)

    v8f acc = {};
    const float* pA = P  + rA * 256 + koff;       // A[m, k0+koff], A[m, k0+koff+1]
    const float* pB = w1 + nB * 256 + koff;       // B[k,n] = w1[n,k]
    for (int k0 = 0; k0 < 256; k0 += 4) {
        v2f a = *(const v2f*)(pA + k0);
        v2f b = *(const v2f*)(pB + k0);
        acc = __builtin_amdgcn_wmma_f32_16x16x4_f32(false, a, false, b,
                                                    (short)0, acc, false, false);
    }
    const float inv   = gamma[nB] * rsqrtf(var[nB] + BN_EPS);
    const float shift = beta[nB] - mean[nB] * inv;
    const int mAdd = (lane >> 4) << 3;            // C/D: M = v + (lane>=16 ? 8 : 0)
#pragma unroll
    for (int v = 0; v < 8; ++v) {
        const int m = Mbase + v + mAdd;
        float val = acc[v] * inv + shift;
        val = 0.5f * val * (1.0f + erff(val * 0.70710678118654752f));  // exact GELU
        Hbuf[m * 64 + nB] = val;
    }
}

// ---------------------------------------------------------------------------
// Kernel 3: S = H @ w2^T + b2.  M=320 (20 tiles), K=64, N=512 (32 tiles).
// ---------------------------------------------------------------------------
__global__ __launch_bounds__(128) void gemm2_kernel(const float* __restrict__ Hbuf,
                                                    const float* __restrict__ w2,
                                                    const float* __restrict__ b2,
                                                    float* __restrict__ S) {
    const int lane  = threadIdx.x & 31;
    const int wv    = threadIdx.x >> 5;
    const int Mbase = blockIdx.x * 16;
    const int Nbase = (blockIdx.y * 4 + wv) * 16;
    const int rA    = Mbase + (lane & 15);
    const int nB    = Nbase + (lane & 15);
    const int koff  = (lane >> 4) << 1;

    v8f acc = {};
    const float* pA = Hbuf + rA * 64 + koff;
    const float* pB = w2   + nB * 64 + koff;
#pragma unroll
    for (int k0 = 0; k0 < 64; k0 += 4) {
        v2f a = *(const v2f*)(pA + k0);
        v2f b = *(const v2f*)(pB + k0);
        acc = __builtin_amdgcn_wmma_f32_16x16x4_f32(false, a, false, b,
                                                    (short)0, acc, false, false);
    }
    const float bias = b2[nB];
    const int mAdd = (lane >> 4) << 3;
#pragma unroll
    for (int v = 0; v < 8; ++v) {
        const int m = Mbase + v + mAdd;
        S[m * 512 + nB] = acc[v] + bias;
    }
}

// ---------------------------------------------------------------------------
// Kernel 4: softmax over G=2 and mix dyn_weight / dyn_bias.
// idx < 73728 -> per-sample 3x3 weights; else -> per-sample bias.
// ---------------------------------------------------------------------------
__global__ __launch_bounds__(256) void combine_kernel(const float* __restrict__ S,
                                                      const float* __restrict__ dynW,
                                                      const float* __restrict__ dynB,
                                                      float* __restrict__ Wbuf,
                                                      float* __restrict__ Bbuf) {
    const int idx = blockIdx.x * 256 + threadIdx.x;
    if (idx < 32 * 256 * 9) {
        const int b   = idx / 2304;
        const int rem = idx % 2304;
        const int c   = rem / 9;
        const int kk  = rem % 9;
        const int row = b * 9 + kk;
        const float s0 = S[row * 512 + c];
        const float s1 = S[row * 512 + 256 + c];
        const float mx = fmaxf(s0, s1);
        const float e0 = expf(s0 - mx), e1 = expf(s1 - mx);
        Wbuf[(b * 256 + c) * 9 + kk] =
            (e0 * dynW[c * 9 + kk] + e1 * dynW[2304 + c * 9 + kk]) / (e0 + e1);
    } else {
        const int j = idx - 32 * 256 * 9;         // j < 8192
        const int b = j >> 8, c = j & 255;
        const int row = 288 + b;
        const float s0 = S[row * 512 + c];
        const float s1 = S[row * 512 + 256 + c];
        const float mx = fmaxf(s0, s1);
        const float e0 = expf(s0 - mx), e1 = expf(s1 - mx);
        Bbuf[j] = (e0 * dynB[c] + e1 * dynB[256 + c]) / (e0 + e1);
    }
}

// ---------------------------------------------------------------------------
// Kernel 5: per-sample depthwise 3x3 conv, pad=1, + per-sample bias.
// One block per (b,c) plane; plane staged in LDS with zero halo (98x98).
// ---------------------------------------------------------------------------
__global__ __launch_bounds__(256) void dwconv_kernel(const float* __restrict__ x,
                                                     const float* __restrict__ Wbuf,
                                                     const float* __restrict__ Bbuf,
                                                     float* __restrict__ y) {
    __shared__ float sm[98 * 98];                 // 38,416 B of the 320 KB WGP LDS
    const int t = threadIdx.x;
    const int plane = blockIdx.x;
    const float* xp = x + (size_t)plane * 9216;

    for (int i = t; i < 98 * 98; i += 256) sm[i] = 0.f;
    __syncthreads();
    for (int g = t; g < 2304; g += 256) {
        const float4 v = ((const float4*)xp)[g];
        const int row = g / 24;
        const int col = (g % 24) * 4;
        float* d = &sm[(row + 1) * 98 + col + 1];
        d[0] = v.x; d[1] = v.y; d[2] = v.z; d[3] = v.w;
    }
    __syncthreads();

    float wr[9];
#pragma unroll
    for (int i = 0; i < 9; ++i) wr[i] = Wbuf[plane * 9 + i];
    const float bias = Bbuf[plane];

    float* yp = y + (size_t)plane * 9216;
    for (int g = t; g < 2304; g += 256) {
        const int orow = g / 24;
        const int ocol = (g % 24) * 4;
        float r[3][6];                            // 3 rows x 6 cols feed 4 outputs
#pragma unroll
        for (int dy = 0; dy < 3; ++dy) {
            const float* src = &sm[(orow + dy) * 98 + ocol];
#pragma unroll
            for (int dx = 0; dx < 6; ++dx) r[dy][dx] = src[dx];
        }
        float o[4];
#pragma unroll
        for (int j = 0; j < 4; ++j) {
            float a = bias;
#pragma unroll
            for (int dy = 0; dy < 3; ++dy)
#pragma unroll
                for (int dx = 0; dx < 3; ++dx)
                    a = fmaf(wr[dy * 3 + dx], r[dy][dx + j], a);
            o[j] = a;
        }
        ((float4*)yp)[g] = make_float4(o[0], o[1], o[2], o[3]);
    }
}

// ---------------------------------------------------------------------------
extern "C" void kernel_launch(void* const* d_in, const int* in_sizes, int n_in,
                              void* d_out, int out_size, void* d_ws, size_t ws_size,
                              hipStream_t stream) {
    (void)in_sizes; (void)n_in; (void)out_size; (void)ws_size;
    const float* x     = (const float*)d_in[0];
    const float* dynW  = (const float*)d_in[1];
    const float* dynB  = (const float*)d_in[2];
    const float* w1    = (const float*)d_in[3];
    const float* gamma = (const float*)d_in[4];
    const float* beta  = (const float*)d_in[5];
    const float* mean  = (const float*)d_in[6];
    const float* var   = (const float*)d_in[7];
    const float* w2    = (const float*)d_in[8];
    const float* b2    = (const float*)d_in[9];
    float* out = (float*)d_out;

    // workspace layout (floats): P[320*256] | H[320*64] | S[320*512] | W[8192*9] | bias[8192]
    float* P  = (float*)d_ws;
    float* Hb = P  + 320 * 256;
    float* S  = Hb + 320 * 64;
    float* Wb = S  + 320 * 512;
    float* Bb = Wb + 8192 * 9;

    pool_kernel   <<<8192,        256, 0, stream>>>(x, P);
    gemm1_kernel  <<<20,          128, 0, stream>>>(P, w1, gamma, beta, mean, var, Hb);
    gemm2_kernel  <<<dim3(20, 8), 128, 0, stream>>>(Hb, w2, b2, S);
    combine_kernel<<<320,         256, 0, stream>>>(S, dynW, dynB, Wb, Bb);
    dwconv_kernel <<<8192,        256, 0, stream>>>(x, Wb, Bb, out);
}